// OneHotEncoding_69320772158123
// MI455X (gfx1250) — compile-verified
//
#include <hip/hip_runtime.h>

// out[t, h] = W[h, seq[t]] + b[h]
//   seq: [32768] int32, W: [1024, 50257] f32 (row-major), b: [1024] f32
//   out: [32768, 1024] f32
//
// Pure memory-bound gather (AI ~ 0.008 FLOP/B); min traffic ~134MB NT stores +
// ~205MB W reads (each 128B line of W reused ~21x across tokens via L2)
// -> ~15us at 23.3 TB/s. WMMA is deliberately absent: dense one-hot GEMM would
// be ~3.4 TFLOP of wasted math, ~100x slower than the gather formulation.
//
// Strategy:
//  - lanes span hidden dim -> coalesced global_store_b128 th:TH_STORE_NT
//    (output streamed once; keep it out of L2 to protect W's working set)
//  - 8 tokens per block: one uniform scalar b256 read of indices, then ALL
//    32 gather loads issued branch-free before any store -> max memory-level
//    parallelism per wave (loads return in order; partial s_wait_loadcnt
//    lets early-token stores overlap in-flight later loads)

#define VOCAB  50257
#define HIDDEN 1024
#define TPB_TOKENS 8   // tokens per block

typedef __attribute__((ext_vector_type(4))) float v4f;

__global__ __launch_bounds__(256) void onehot_gather_bias_kernel(
    const int* __restrict__ seq,
    const float* __restrict__ W,
    const float* __restrict__ b,
    float* __restrict__ out,
    int n_tokens)
{
    const int tid = threadIdx.x;          // 256 threads
    const int h0  = tid << 2;             // each thread owns rows h0..h0+3

    // Bias for this thread's 4 rows: one b128 load, L2-resident after warmup.
    const v4f bias = *reinterpret_cast<const v4f*>(b + h0);

    // Base of this thread's 4 W rows (row stride = VOCAB floats).
    const float* __restrict__ Wr = W + (size_t)h0 * VOCAB;

    const int tok0 = blockIdx.x * TPB_TOKENS;

    if (tok0 + TPB_TOKENS <= n_tokens) {
        // ---- fast path: full block, no per-token branches ----

        // Uniform batch read of all 8 indices (scalar path, single KMcnt wait).
        int idx[TPB_TOKENS];
        #pragma unroll
        for (int t = 0; t < TPB_TOKENS; ++t) idx[t] = seq[tok0 + t];

        // Issue all 32 random 4B gathers back-to-back (default temporal hint:
        // pulled 128B lines stay in L2 for cross-token reuse).
        v4f v[TPB_TOKENS];
        #pragma unroll
        for (int t = 0; t < TPB_TOKENS; ++t) {
            const size_t o = (size_t)idx[t];
            v[t].x = Wr[o];
            v[t].y = Wr[o + VOCAB];
            v[t].z = Wr[o + 2 * (size_t)VOCAB];
            v[t].w = Wr[o + 3 * (size_t)VOCAB];
        }

        // Drain in order: bias add + coalesced NT b128 stores.
        #pragma unroll
        for (int t = 0; t < TPB_TOKENS; ++t) {
            v4f r = v[t];
            r.x += bias.x;
            r.y += bias.y;
            r.z += bias.z;
            r.w += bias.w;
            v4f* dst = reinterpret_cast<v4f*>(out + (size_t)(tok0 + t) * HIDDEN + h0);
            __builtin_nontemporal_store(r, dst);
        }
    } else {
        // ---- tail path (never taken for 32768 tokens, kept for correctness) ----
        for (int t = 0; t < TPB_TOKENS; ++t) {
            const int tok = tok0 + t;
            if (tok >= n_tokens) return;
            const int idx = seq[tok];
            const size_t o = (size_t)idx;
            v4f r;
            r.x = Wr[o]                       + bias.x;
            r.y = Wr[o + VOCAB]               + bias.y;
            r.z = Wr[o + 2 * (size_t)VOCAB]   + bias.z;
            r.w = Wr[o + 3 * (size_t)VOCAB]   + bias.w;
            v4f* dst = reinterpret_cast<v4f*>(out + (size_t)tok * HIDDEN + h0);
            __builtin_nontemporal_store(r, dst);
        }
    }
}

extern "C" void kernel_launch(void* const* d_in, const int* in_sizes, int n_in,
                              void* d_out, int out_size, void* d_ws, size_t ws_size,
                              hipStream_t stream) {
    const int*   seq = (const int*)  d_in[0];   // [B*S] int32
    const float* W   = (const float*)d_in[1];   // [HIDDEN, VOCAB] f32
    const float* b   = (const float*)d_in[2];   // [HIDDEN] f32
    float*       out = (float*)d_out;           // [B*S, HIDDEN] f32

    const int n_tokens = in_sizes[0];
    const int blocks   = (n_tokens + TPB_TOKENS - 1) / TPB_TOKENS;

    onehot_gather_bias_kernel<<<blocks, 256, 0, stream>>>(seq, W, b, out, n_tokens);
}